// Pair_emb_wo_templ_30193620091043
// MI455X (gfx1250) — compile-verified
//
#include <hip/hip_runtime.h>
#include <hip/hip_bf16.h>
#include <math.h>

typedef __attribute__((ext_vector_type(2))) float v2f;
typedef __attribute__((ext_vector_type(4))) float v4f;
typedef __attribute__((ext_vector_type(8))) float v8f;

// ---------------------------------------------------------------------------
// Kernel 1: PLPR[32,256] = padA[32,64] @ Bmat[64,256] via V_WMMA_F32_16X16X4_F32
//   padA[s,c]  = (s < 21) ? emb[s,c] : 0
//   Bmat[k,n]  = (n < 128) ? W[n,k] : W[n-128, 64+k]
//   => PLPR[s, 0:128]   = PL[s] = W_left  @ emb[s]
//      PLPR[s, 128:256] = PR[s] = W_right @ emb[s]
// Grid: 32 blocks x 32 threads (1 wave each). Block -> one 16x16 D tile.
// ---------------------------------------------------------------------------
__global__ __launch_bounds__(32) void plpr_wmma_kernel(
    const float* __restrict__ emb,   // [21,64]
    const float* __restrict__ W,     // [128,129]
    float* __restrict__ plpr)        // [32,256] (workspace)
{
    const int lane  = threadIdx.x;            // 0..31
    const int Mb    = (blockIdx.x & 1) * 16;  // M tile base (0 or 16)
    const int Nb    = (blockIdx.x >> 1) * 16; // N tile base (0..240)
    const int lo    = lane & 15;
    const int khalf = (lane >> 4) * 2;        // lanes 0-15 -> K+{0,1}, 16-31 -> K+{2,3}

    const int mrow = Mb + lo;                 // A-matrix row for this lane
    const int ncol = Nb + lo;                 // B/D column for this lane
    const int wrow = (ncol < 128) ? ncol : (ncol - 128);
    const int coff = (ncol < 128) ? 0 : 64;

    v8f acc = {};
    #pragma unroll
    for (int kc = 0; kc < 16; ++kc) {
        const int k0 = kc * 4 + khalf;
        v2f a, bmat;
        if (mrow < 21) {
            a.x = emb[mrow * 64 + k0];
            a.y = emb[mrow * 64 + k0 + 1];
        } else {
            a.x = 0.0f; a.y = 0.0f;
        }
        bmat.x = W[wrow * 129 + coff + k0];
        bmat.y = W[wrow * 129 + coff + k0 + 1];
        // 8 args: (neg_a, A, neg_b, B, c_mod, C, reuse_a, reuse_b)
        acc = __builtin_amdgcn_wmma_f32_16x16x4_f32(
            false, a, false, bmat, (short)0, acc, false, false);
    }

    // D layout: VGPR r, lanes 0-15 -> M=r, lanes 16-31 -> M=r+8; N = lane&15
    const int mhi = (lane >> 4) * 8;
    #pragma unroll
    for (int r = 0; r < 8; ++r) {
        plpr[(Mb + mhi + r) * 256 + ncol] = acc[r];
    }
}

// ---------------------------------------------------------------------------
// Kernel 2: streaming elementwise pass (store-bandwidth bound).
//   out[b,i,j,d] = (PL[s_i][d] + bias[d]) + PR[s_j][d] + wsep[d]*log(|di-dj|+1)
// One block per (b,i). 256 threads = 8 waves; each wave emits one full
// 128-float j-row (512B contiguous global_store_b128, non-temporal).
// ---------------------------------------------------------------------------
#define MAX_L 1024

__global__ __launch_bounds__(256) void pair_row_kernel(
    const int*   __restrict__ seq,    // [B,L]
    const int*   __restrict__ idx,    // [B,L]
    const float* __restrict__ W,      // [128,129] (for wsep column)
    const float* __restrict__ bias,   // [128]
    const float* __restrict__ plpr,   // [32,256] from kernel 1
    float* __restrict__ out,          // [B,L,L,128]
    int L)
{
    __shared__ float ldsPR[21 * 128];
    __shared__ int   ldsSeq[MAX_L];
    __shared__ int   ldsIdx[MAX_L];

    const int tid = threadIdx.x;
    const int bi  = blockIdx.x;      // = b*L + i
    const int bb  = bi / L;

    // Stage PR table (21x128) and this batch's seq/idx rows into LDS.
    for (int t = tid; t < 21 * 128; t += 256) {
        const int s = t >> 7, d = t & 127;
        ldsPR[t] = plpr[s * 256 + 128 + d];
    }
    for (int t = tid; t < L; t += 256) {
        ldsSeq[t] = seq[bb * L + t];
        ldsIdx[t] = idx[bb * L + t];
    }
    __syncthreads();

    const int d0 = (tid & 31) * 4;   // 4 output channels per lane
    const int jg = tid >> 5;         // wave id -> j phase (uniform per wave)

    const int s_i   = seq[bi];
    const int idx_i = idx[bi];

    // PL row (+bias folded) and wsep column, per-lane registers.
    const float pl0 = plpr[s_i * 256 + d0 + 0] + bias[d0 + 0];
    const float pl1 = plpr[s_i * 256 + d0 + 1] + bias[d0 + 1];
    const float pl2 = plpr[s_i * 256 + d0 + 2] + bias[d0 + 2];
    const float pl3 = plpr[s_i * 256 + d0 + 3] + bias[d0 + 3];
    const float ws0 = W[(d0 + 0) * 129 + 128];
    const float ws1 = W[(d0 + 1) * 129 + 128];
    const float ws2 = W[(d0 + 2) * 129 + 128];
    const float ws3 = W[(d0 + 3) * 129 + 128];

    float* __restrict__ outrow = out + (size_t)bi * (size_t)L * 128u;

    for (int j = jg; j < L; j += 8) {
        const int   dj  = idx_i - ldsIdx[j];
        const float sep = logf(fabsf((float)dj) + 1.0f);
        const float* __restrict__ pr = &ldsPR[ldsSeq[j] * 128 + d0];
        v4f v;
        v.x = pl0 + pr[0] + ws0 * sep;
        v.y = pl1 + pr[1] + ws1 * sep;
        v.z = pl2 + pr[2] + ws2 * sep;
        v.w = pl3 + pr[3] + ws3 * sep;
        // Streaming output (1.07 GB, never re-read): non-temporal b128 store.
        __builtin_nontemporal_store(v, (v4f*)(outrow + (size_t)j * 128u + d0));
    }
}

extern "C" void kernel_launch(void* const* d_in, const int* in_sizes, int n_in,
                              void* d_out, int out_size, void* d_ws, size_t ws_size,
                              hipStream_t stream) {
    const int*   seq  = (const int*)d_in[0];     // [B,L]
    const int*   idx  = (const int*)d_in[1];     // [B,L]
    const float* emb  = (const float*)d_in[2];   // [21,64]
    const float* W    = (const float*)d_in[3];   // [128,129]
    const float* bias = (const float*)d_in[4];   // [128]
    float*       out  = (float*)d_out;           // [B,L,L,128]

    const int N = in_sizes[0];                   // B*L
    const int L = (int)((long long)out_size / ((long long)N * 128)); // out = N*L*128

    float* plpr = (float*)d_ws;                  // 32*256*4 = 32 KB scratch

    // Kernel 1: tiny f32 WMMA GEMM -> PLPR table.
    plpr_wmma_kernel<<<dim3(32), dim3(32), 0, stream>>>(emb, W, plpr);

    // Kernel 2: bandwidth-bound streaming pass.
    pair_row_kernel<<<dim3(N), dim3(256), 0, stream>>>(seq, idx, W, bias, plpr, out, L);
}